// Perceptron_86732569575514
// MI455X (gfx1250) — compile-verified
//
#include <hip/hip_runtime.h>

#define NW 10
#define HID 32
#define NF 128
#define BATCH 65536
#define BTILE 32
#define LN_EPS 1e-5f
#define NORM_EPS 1e-12f

typedef __attribute__((ext_vector_type(2))) float v2f;
typedef __attribute__((ext_vector_type(8))) float v8f;

#if defined(__has_builtin)
#  if __has_builtin(__builtin_amdgcn_rcpf)
#    define FRCP(x) __builtin_amdgcn_rcpf(x)
#  endif
#  if __has_builtin(__builtin_amdgcn_rsqf)
#    define FRSQ(x) __builtin_amdgcn_rsqf(x)
#  endif
#  if __has_builtin(__builtin_amdgcn_sqrtf)
#    define FSQRT(x) __builtin_amdgcn_sqrtf(x)
#  endif
#endif
#ifndef FRCP
#  define FRCP(x) (1.0f / (x))
#endif
#ifndef FRSQ
#  define FRSQ(x) rsqrtf(x)
#endif
#ifndef FSQRT
#  define FSQRT(x) sqrtf(x)
#endif

__global__ __launch_bounds__(256)
void perceptron_fused(const float* __restrict__ src, const float* __restrict__ Wi,
                      const float* __restrict__ bi,  const float* __restrict__ pos,
                      const float* __restrict__ emb, const float* __restrict__ ln_g,
                      const float* __restrict__ ln_b, const float* __restrict__ Wo,
                      const float* __restrict__ bo,  const float* __restrict__ Wo2,
                      const float* __restrict__ bo2, float* __restrict__ out)
{
  __shared__ __align__(16) float sX[NW][BTILE][HID];   // hidden activations
  __shared__ __align__(16) float sXC[BTILE][HID];      // Wo2-combined hidden
  __shared__ __align__(16) float sQn[2][NW][HID];      // normalized queries
  __shared__ __align__(16) float sBP[NW][HID];         // bi + pos fused bias

  const int tid   = threadIdx.x;
  const int wv    = tid >> 5;
  const int lane  = tid & 31;
  const int hv    = lane >> 4;     // lane half (WMMA layout)
  const int lr    = lane & 15;
  const int bBase = blockIdx.x * BTILE;

  // ---------------- phase 0: tiny parameter prep ----------------
  if (tid < NW * HID) {
    int n = tid >> 5, h = tid & 31;
    sBP[n][h] = bi[h] + pos[n * HID + h];
  }
  if (tid < 2 * NW) {
    const float* e = emb + tid * HID;
    float ss = 0.f;
    for (int h = 0; h < HID; ++h) ss += e[h] * e[h];
    float inv = FRCP(fmaxf(FSQRT(ss), NORM_EPS));
    float* q = &sQn[0][0][0] + tid * HID;
    for (int h = 0; h < HID; ++h) q[h] = e[h] * inv;
  }
  __syncthreads();

  // ---------------- phase 1: input projection via V_WMMA_F32_16X16X4_F32 ----------------
  // x[n,b,h] = src[n,b,:].Wi[h,:] + bi[h] + pos[n,h]
  // 32-bit A 16x4 layout: lanes 0-15 row=lr K{0,1}; lanes 16-31 row=lr K{2,3}
  // 32-bit B 4x16 layout: lanes 0-15 col=lr K{0,1}; lanes 16-31 col=lr K{2,3}
  // Fragments are two consecutive f32 per lane: pure b64 loads, no converts.
  // A fragments are loaded into a 32-entry array FIRST (32 loads in flight,
  // clause-able) and only then consumed by the chained WMMAs -> the scheduler
  // cannot collapse them into one serialized load/wait/wmma register pair.
  {
    const int bsub  = wv & 1;         // which 16-row batch sub-tile
    const int nt    = (wv >> 1) & 1;  // which 16-col hidden tile
    const int ngrp  = wv >> 2;        // which half of the window
    const int hBase = nt * 16;
    const int bb    = bsub * 16;

    // B fragments (Wi^T): B[k, nh] = Wi[hBase+nh][4*kc + k]; cached once per wave.
    v2f bfrag[32];
    {
      const float* wrow = Wi + (size_t)(hBase + lr) * NF + 2 * hv;
      #pragma unroll
      for (int kc = 0; kc < 32; ++kc)
        bfrag[kc] = *(const v2f*)(wrow + 4 * kc);
    }

    for (int ni = 0; ni < 5; ++ni) {
      const int n = ngrp * 5 + ni;
      const float* arow = src + ((size_t)n * BATCH + bBase + bb + lr) * NF + 2 * hv;
      if (ni + 1 < 5) __builtin_prefetch(arow + (size_t)BATCH * NF, 0, 3);

      // batch all K fragments of this tile: 32 independent b64 loads in flight
      v2f afrag[32];
      #pragma unroll
      for (int kc = 0; kc < 32; ++kc)
        afrag[kc] = *(const v2f*)(arow + 4 * kc);

      v8f acc;
      const float cinit = sBP[n][hBase + lr];   // same for every row of the tile
      #pragma unroll
      for (int j = 0; j < 8; ++j) acc[j] = cinit;

      #pragma unroll
      for (int kc = 0; kc < 32; ++kc)
        acc = __builtin_amdgcn_wmma_f32_16x16x4_f32(false, afrag[kc], false, bfrag[kc],
                                                    (short)0, acc, false, false);

      // C/D layout: VGPR j -> row M=j (lanes 0-15) / M=j+8 (lanes 16-31), col N=lr
      #pragma unroll
      for (int j = 0; j < 8; ++j)
        sX[n][bb + j + 8 * hv][hBase + lr] = acc[j];
    }
  }
  __syncthreads();

  // ---------------- phase 2: two cosine-attention layers ----------------
  // 8 lanes cooperate per batch element (4 hidden each); shfl_xor butterflies.
  {
    const int bl = tid >> 3;         // batch element in tile
    const int hs = (tid & 7) * 4;    // hidden slice
    float xr[NW][4], xn[NW][4];
    #pragma unroll
    for (int m = 0; m < NW; ++m) {
      const float4 v = *(const float4*)&sX[m][bl][hs];
      xr[m][0] = v.x; xr[m][1] = v.y; xr[m][2] = v.z; xr[m][3] = v.w;
    }
    #pragma unroll
    for (int layer = 0; layer < 2; ++layer) {
      float g[4], bt[4];
      #pragma unroll
      for (int j = 0; j < 4; ++j) {
        g[j]  = ln_g[layer * HID + hs + j];
        bt[j] = ln_b[layer * HID + hs + j];
      }
      float inv[NW];
      #pragma unroll
      for (int m = 0; m < NW; ++m) {
        float ss = xr[m][0]*xr[m][0] + xr[m][1]*xr[m][1]
                 + xr[m][2]*xr[m][2] + xr[m][3]*xr[m][3];
        ss += __shfl_xor(ss, 1); ss += __shfl_xor(ss, 2); ss += __shfl_xor(ss, 4);
        inv[m] = FRCP(fmaxf(FSQRT(ss), NORM_EPS));
      }
      for (int n = 0; n < NW; ++n) {
        const float q0 = sQn[layer][n][hs],     q1 = sQn[layer][n][hs + 1];
        const float q2 = sQn[layer][n][hs + 2], q3 = sQn[layer][n][hs + 3];
        // cosine scores are bounded to [-1,1]: softmax without max-shift is safe
        // and mathematically identical to the reference.
        float sc[NW]; float se = 0.f;
        #pragma unroll
        for (int m = 0; m < NW; ++m) {
          float d = q0*xr[m][0] + q1*xr[m][1] + q2*xr[m][2] + q3*xr[m][3];
          d += __shfl_xor(d, 1); d += __shfl_xor(d, 2); d += __shfl_xor(d, 4);
          sc[m] = __expf(d * inv[m]);
          se += sc[m];
        }
        const float rs = FRCP(se);
        float o0 = 0.f, o1 = 0.f, o2 = 0.f, o3 = 0.f;
        #pragma unroll
        for (int m = 0; m < NW; ++m) {
          const float w = sc[m];
          o0 += w * xr[m][0]; o1 += w * xr[m][1]; o2 += w * xr[m][2]; o3 += w * xr[m][3];
        }
        o0 = o0 * rs + xr[n][0]; o1 = o1 * rs + xr[n][1];
        o2 = o2 * rs + xr[n][2]; o3 = o3 * rs + xr[n][3];
        float s = o0 + o1 + o2 + o3;
        s += __shfl_xor(s, 1); s += __shfl_xor(s, 2); s += __shfl_xor(s, 4);
        const float mu = s * (1.0f / HID);
        const float d0 = o0 - mu, d1 = o1 - mu, d2 = o2 - mu, d3 = o3 - mu;
        float va = d0*d0 + d1*d1 + d2*d2 + d3*d3;
        va += __shfl_xor(va, 1); va += __shfl_xor(va, 2); va += __shfl_xor(va, 4);
        const float r = FRSQ(va * (1.0f / HID) + LN_EPS);
        xn[n][0] = d0*r*g[0] + bt[0]; xn[n][1] = d1*r*g[1] + bt[1];
        xn[n][2] = d2*r*g[2] + bt[2]; xn[n][3] = d3*r*g[3] + bt[3];
      }
      #pragma unroll
      for (int n = 0; n < NW; ++n) {
        xr[n][0] = xn[n][0]; xr[n][1] = xn[n][1]; xr[n][2] = xn[n][2]; xr[n][3] = xn[n][3];
      }
    }
    // fold the window combine (Wo2) BEFORE the output GEMM (linear, so legal)
    float c0 = 0.f, c1 = 0.f, c2 = 0.f, c3 = 0.f;
    #pragma unroll
    for (int n = 0; n < NW; ++n) {
      const float w2 = Wo2[n];
      c0 += w2*xr[n][0]; c1 += w2*xr[n][1]; c2 += w2*xr[n][2]; c3 += w2*xr[n][3];
    }
    float4 cc; cc.x = c0; cc.y = c1; cc.z = c2; cc.w = c3;
    *(float4*)&sXC[bl][hs] = cc;
  }
  __syncthreads();

  // ---------------- phase 3: output projection via V_WMMA_F32_16X16X4_F32 ----------------
  // out[b,f] = xc[b,:].Wo[f,:] + bo[f]*sum(Wo2) + bo2   (K = 32 -> 8 chained WMMAs)
  {
    const int mt = wv & 1;
    float sumw2 = 0.f;
    #pragma unroll
    for (int n = 0; n < NW; ++n) sumw2 += Wo2[n];
    const float bo2v = bo2[0];

    // A fragments from LDS: rows = mt*16 + lr, K = hidden
    v2f afrag[8];
    {
      const float* xrow = &sXC[mt * 16 + lr][2 * hv];
      #pragma unroll
      for (int kc = 0; kc < 8; ++kc)
        afrag[kc] = *(const v2f*)(xrow + 4 * kc);
    }
    #pragma unroll
    for (int t = 0; t < 2; ++t) {
      const int nt = (wv >> 1) + t * 4;
      const int f  = nt * 16 + lr;
      const float* wrow = Wo + (size_t)f * HID + 2 * hv;

      // preload all B fragments before the WMMA chain (keep loads in flight)
      v2f bfr[8];
      #pragma unroll
      for (int kc = 0; kc < 8; ++kc)
        bfr[kc] = *(const v2f*)(wrow + 4 * kc);

      v8f acc;
      const float cb = bo[f] * sumw2 + bo2v;
      #pragma unroll
      for (int j = 0; j < 8; ++j) acc[j] = cb;
      #pragma unroll
      for (int kc = 0; kc < 8; ++kc)
        acc = __builtin_amdgcn_wmma_f32_16x16x4_f32(false, afrag[kc], false, bfr[kc],
                                                    (short)0, acc, false, false);

      float* orow = out + (size_t)(bBase + mt * 16 + 8 * hv) * NF + f;
      #pragma unroll
      for (int j = 0; j < 8; ++j)
        orow[(size_t)j * NF] = acc[j];
    }
  }
}

extern "C" void kernel_launch(void* const* d_in, const int* in_sizes, int n_in,
                              void* d_out, int out_size, void* d_ws, size_t ws_size,
                              hipStream_t stream) {
  const float* src  = (const float*)d_in[0];
  const float* Wi   = (const float*)d_in[1];
  const float* bi   = (const float*)d_in[2];
  const float* pos  = (const float*)d_in[3];
  const float* emb  = (const float*)d_in[4];
  const float* ln_g = (const float*)d_in[5];
  const float* ln_b = (const float*)d_in[6];
  const float* Wo   = (const float*)d_in[7];
  const float* bo   = (const float*)d_in[8];
  const float* Wo2  = (const float*)d_in[9];
  const float* bo2  = (const float*)d_in[10];
  float* out = (float*)d_out;

  dim3 grid(BATCH / BTILE), block(256);
  hipLaunchKernelGGL(perceptron_fused, grid, block, 0, stream,
                     src, Wi, bi, pos, emb, ln_g, ln_b, Wo, bo, Wo2, bo2, out);
}